// PermutationSinkhornLayer_56762287784053
// MI455X (gfx1250) — compile-verified
//
#include <hip/hip_runtime.h>
#include <hip/hip_bf16.h>

// ---------------------------------------------------------------------------
// PermutationSinkhornLayer for MI455X (gfx1250, wave32, WMMA + TDM)
//   B=32, S=256, D=512, HEADS=8, hd=64, AR=4, TEMP=1, NEG=-999999
//   out = [sample_permu(128,256,256), masked_refinement_acti(128,256,256),
//          kl(128,1), log_nominator(128,1), log_normalize_const(128,1)]
// ---------------------------------------------------------------------------

#define SB      32
#define SS      256
#define SD      512
#define SHEADS  8
#define SHD     64
#define SAR     4
#define NEGC    (-999999.0f)
#define PITCH   257          // 257 % 64 == 1 -> conflict-free row & col passes
#define SINK_ITERS  200
#define BETHE_ITERS 30

typedef __attribute__((ext_vector_type(2))) float v2f;
typedef __attribute__((ext_vector_type(8))) float v8f;
typedef unsigned int u32x4 __attribute__((ext_vector_type(4)));
typedef int          i32x4 __attribute__((ext_vector_type(4)));
typedef int          i32x8 __attribute__((ext_vector_type(8)));

#if defined(__HIP_DEVICE_COMPILE__)
#if !__has_builtin(__builtin_amdgcn_wmma_f32_16x16x4_f32)
#error "gfx1250 f32 WMMA builtin missing - switch to 16x16x32 f16 path"
#endif
#endif

#if __has_builtin(__builtin_amdgcn_tensor_load_to_lds) && \
    __has_builtin(__builtin_amdgcn_s_wait_tensorcnt)
#define HAVE_TDM 1
#else
#define HAVE_TDM 0
#endif

__device__ __forceinline__ v8f wmma_f32(v2f a, v2f b, v8f c) {
  // D = A(16x4,f32) * B(4x16,f32) + C(16x16,f32)
  return __builtin_amdgcn_wmma_f32_16x16x4_f32(
      /*neg_a=*/false, a, /*neg_b=*/false, b,
      /*c_mod=*/(short)0, c, /*reuse_a=*/false, /*reuse_b=*/false);
}

#if HAVE_TDM
// TDM: DMA a 256x256 f32 row-major tile from global into LDS, inserting one
// pad DWORD after every 256 DWORDs -> row pitch 257 floats (matches PITCH).
// Descriptor per CDNA5 ISA sec.8: group0 = count/lds_addr/global_addr/type,
// group1 = data_size + pad config + dims/strides. Groups 2/3 = 0 (2-D tile).
// This toolchain exposes the 6-arg builtin form:
//   (uint32x4 g0, int32x8 g1, int32x4, int32x4, int32x8, i32 cpol)
__device__ __forceinline__ void tdm_load_256x256_padded(const float* src,
                                                        unsigned lds_byte_off) {
  const unsigned long long ga = (unsigned long long)src;
  u32x4 g0;
  g0[0] = 1u;                                              // count=1, user mode
  g0[1] = lds_byte_off;                                    // lds_addr
  g0[2] = (unsigned)ga;                                    // global_addr[31:0]
  g0[3] = (unsigned)((ga >> 32) & 0x01FFFFFFu)             // global_addr[56:32]
        | 0x80000000u;                                     // type=2 ("image")
  i32x8 g1;
  g1[0] = (2 << 16)        // data_size = 4 bytes
        | (1 << 20)        // pad_enable
        | (7 << 22)        // pad_interval: 256 DWORDs between pads
        | (0 << 25);       // pad_amount: 1 DWORD  -> pitch 257
  g1[1] = (int)(256u << 16);   // tensor_dim0 = 256 (bits 79:48, low half)
  g1[2] = (int)(256u << 16);   // tensor_dim0 hi = 0 | tensor_dim1 = 256 (low)
  g1[3] = (int)(256u << 16);   // tensor_dim1 hi = 0 | tile_dim0 = 256
  g1[4] = 256;                 // tile_dim1 = 256, tile_dim2 = 0
  g1[5] = 256;                 // tensor_dim0_stride = 256 (low 32)
  g1[6] = 0;                   // stride hi | tensor_dim1_stride low
  g1[7] = 0;
  const i32x4 z4 = {0, 0, 0, 0};
  const i32x8 z8 = {0, 0, 0, 0, 0, 0, 0, 0};
  __builtin_amdgcn_tensor_load_to_lds(g0, g1, z4, z4, z8, /*cpol=*/0);
}
#endif

// ------------------------- GEMM: C = act(A*W + bias) -----------------------
// One 16x16 output tile per wave. A: MxK row-major, W: KxN row-major.
__global__ __launch_bounds__(256) void gemm_bias_act(
    const float* __restrict__ A, const float* __restrict__ W,
    const float* __restrict__ bias, float* __restrict__ C,
    int M, int N, int K, int do_relu)
{
  const int wave = threadIdx.x >> 5;
  const int lane = threadIdx.x & 31;
  const int tile = blockIdx.x * 8 + wave;
  const int tilesN = N >> 4;
  const int tm = tile / tilesN;
  const int tn = tile - tm * tilesN;
  if (tm * 16 >= M) return;

  const int half = lane >> 4;      // 0: lanes 0-15, 1: lanes 16-31
  const int l16  = lane & 15;
  const int kb   = half << 1;      // K sub-offset {0,2}

  const float* Arow = A + (size_t)(tm * 16 + l16) * K;   // this lane's A row
  const float* Wcol = W + tn * 16 + l16;                 // this lane's B col

  v8f acc = {};
  for (int k = 0; k < K; k += 4) {
    if ((k & 31) == 0 && k + 64 < K)
      __builtin_prefetch(Arow + k + 64, 0, 0);           // global_prefetch_b8
    v2f a = *(const v2f*)(Arow + k + kb);                // A[m][k+kb..k+kb+1]
    v2f b;
    b.x = Wcol[(size_t)(k + kb)     * N];                // B[k+kb  ][n]
    b.y = Wcol[(size_t)(k + kb + 1) * N];                // B[k+kb+1][n]
    acc = wmma_f32(a, b, acc);
  }

  const int col  = tn * 16 + l16;
  const float bv = bias[col];
  const int row0 = tm * 16 + (half << 3);
#pragma unroll
  for (int v = 0; v < 8; ++v) {
    float x = acc[v] + bv;
    if (do_relu) x = fmaxf(x, 0.0f);
    C[(size_t)(row0 + v) * N + col] = x;
  }
}

// ------------- scores: activations[b,i,j] = sum_h q.k^T/8, masked ----------
__global__ __launch_bounds__(256) void scores_kernel(
    const float* __restrict__ acti,        // (B*S, 1024): per head [q(64)|k(64)]
    const int* __restrict__ qmask,         // (B, S)
    const int* __restrict__ pretrain_done, // scalar
    float* __restrict__ act_out)           // (B, S, S)
{
  const int wave = threadIdx.x >> 5;
  const int lane = threadIdx.x & 31;
  const int tile = blockIdx.x * 8 + wave;  // 32 batches * 16 * 16 tiles = 8192
  const int b  = tile >> 8;
  const int t2 = tile & 255;
  const int ti = t2 >> 4, tj = t2 & 15;

  const int half = lane >> 4;
  const int l16  = lane & 15;
  const int kb   = half << 1;

  const int N2 = 2 * SD;                  // 1024
  const float* qrow = acti + ((size_t)b * SS + ti * 16 + l16) * N2;        // q row i
  const float* krow = acti + ((size_t)b * SS + tj * 16 + l16) * N2 + SHD;  // k row j

  v8f acc = {};
  for (int h = 0; h < SHEADS; ++h) {
    const float* qh = qrow + h * (2 * SHD);
    const float* kh = krow + h * (2 * SHD);
    for (int c = 0; c < SHD; c += 4) {
      v2f a  = *(const v2f*)(qh + c + kb);   // A[i][c+kb..]
      v2f bb = *(const v2f*)(kh + c + kb);   // B[c+kb][j] = k[j][c+kb]
      acc = wmma_f32(a, bb, acc);
    }
  }

  const int pd = *pretrain_done;
  const int j  = tj * 16 + l16;
  const int mj = qmask[b * SS + j] != 0;
  const int i0 = ti * 16 + (half << 3);
#pragma unroll
  for (int v = 0; v < 8; ++v) {
    const int i  = i0 + v;
    const int mi = qmask[b * SS + i] != 0;
    float val;
    if (mi && mj) {
      val = acc[v] * 0.125f;         // 1/sqrt(64)
      if (!pd) val = 0.0f;           // activations -= activations*vmat
    } else {
      val = 8.0f * NEGC;             // all 8 heads masked to NEG, then summed
      if (i == j) val = 0.0f;        // invalid-diagonal zeroing (i==j -> mi==mj==0)
    }
    act_out[((size_t)b * SS + i) * SS + j] = val;
  }
}

// ------------------------------ reductions ---------------------------------
__device__ __forceinline__ float block_sum(float v, float* red, int t) {
  red[t] = v; __syncthreads();
  for (int off = 128; off > 0; off >>= 1) {
    if (t < off) red[t] += red[t + off];
    __syncthreads();
  }
  float r = red[0]; __syncthreads();
  return r;
}

// kl[b] = -nsq + sum(masked) + sum_valid exp(-max(masked,-20))   (TEMP=1)
__global__ __launch_bounds__(256) void kl_kernel(
    const float* __restrict__ act, const int* __restrict__ qmask,
    float* __restrict__ out_kl)   // (B*AR)
{
  __shared__ float red[256];
  const int b = blockIdx.x, t = threadIdx.x;
  float nv = (float)(qmask[b * SS + t] != 0);
  float s1 = 0.f, s2 = 0.f;
  const float* A = act + (size_t)b * SS * SS;
  for (int idx = t; idx < SS * SS; idx += 256) {
    const int i = idx >> 8, j = idx & 255;
    if ((qmask[b * SS + i] != 0) && (qmask[b * SS + j] != 0)) {
      const float m = A[idx];
      s1 += m;
      s2 += __expf(-fmaxf(m, -20.0f));
    }
  }
  const float nvt = block_sum(nv, red, t);
  const float s1t = block_sum(s1, red, t);
  const float s2t = block_sum(s2, red, t);
  if (t == 0) {
    const float kl = -(nvt * nvt) + s1t + s2t;   // lgamma(2)=0, TEMP=1
#pragma unroll
    for (int r = 0; r < SAR; ++r) out_kl[b * SAR + r] = kl;
  }
}

// ------------- one Sinkhorn-log round over LDS-resident matrix -------------
__device__ __forceinline__ void sinkhorn_norm(float* L, int t) {
  // Row pass: thread t owns row t (pitch 257 -> conflict-free)
  float* row = L + t * PITCH;
  float m = -1e38f;
  for (int j = 0; j < SS; ++j) m = fmaxf(m, row[j]);
  float s = 0.f;
  for (int j = 0; j < SS; ++j) s += __expf(row[j] - m);
  const float lr = m + __logf(s);
  for (int j = 0; j < SS; ++j) row[j] -= lr;
  __syncthreads();
  // Column pass: thread t owns column t
  m = -1e38f;
  for (int i = 0; i < SS; ++i) m = fmaxf(m, L[i * PITCH + t]);
  s = 0.f;
  for (int i = 0; i < SS; ++i) s += __expf(L[i * PITCH + t] - m);
  const float lc = m + __logf(s);
  for (int i = 0; i < SS; ++i) L[i * PITCH + t] -= lc;
  __syncthreads();
}

// --------------- Bethe free-energy: log_normalize_const --------------------
__global__ __launch_bounds__(256) void bethe_kernel(
    const float* __restrict__ act, const int* __restrict__ qmask,
    float* __restrict__ out_lnc)  // (B*AR)
{
  extern __shared__ float sm[];
  float* L   = sm;                 // 256*257 floats, LDS-resident
  float* red = sm + SS * PITCH;    // 256
  const int b = blockIdx.x, t = threadIdx.x;
  const float* logw = act + (size_t)b * SS * SS;

#if HAVE_TDM
  // One TDM DMA stages the whole 256KB tile into LDS with hardware pitch-257
  // padding; zero VGPR traffic. TDM ignores EXEC, so gate to wave 0.
  if (threadIdx.x < 32) {
    tdm_load_256x256_padded(logw, __builtin_amdgcn_groupstaticsize());
    __builtin_amdgcn_s_wait_tensorcnt(0);
  }
  __syncthreads();
#else
  for (int i = 0; i < SS; ++i) L[i * PITCH + t] = logw[i * SS + t];
  __syncthreads();
#endif

  sinkhorn_norm(L, t);                         // initial Bm = exp(sinkhorn(logw,1))
  for (int it = 0; it < BETHE_ITERS; ++it) {
    for (int i = 0; i < SS; ++i) {
      float e = __expf(L[i * PITCH + t]);
      e = fminf(fmaxf(e, 0.0f), 1.0f - 1e-6f);
      L[i * PITCH + t] = logw[i * SS + t] + log1pf(-e);
    }
    __syncthreads();
    sinkhorn_norm(L, t);
  }

  // t1 + t2 + t3 with mba = exp(L)*vmat, masked = logw*vmat, m1 = (1-mba)*vmat
  const int mt = qmask[b * SS + t] != 0;       // column-t mask
  float acc = 0.f;
  for (int i = 0; i < SS; ++i) {
    const int vm = (qmask[b * SS + i] != 0) && mt;
    if (vm) {
      const float Bm = __expf(L[i * PITCH + t]);
      const float mk = logw[i * SS + t];
      const float m1 = 1.0f - Bm;
      acc += Bm * mk
           - Bm * __logf(Bm + 1e-10f)
           + m1 * __logf(m1 + 1e-10f);
    }
  }
  const float total = block_sum(acc, red, t);
  if (t == 0) {
#pragma unroll
    for (int r = 0; r < SAR; ++r) out_lnc[b * SAR + r] = total;
  }
}

// --------- Sinkhorn(200) + greedy assignment + outputs, 1 WG / matrix ------
__global__ __launch_bounds__(256) void sinkhorn_kernel(
    const float* __restrict__ act,      // (B,S,S)
    const float* __restrict__ gumbel,   // (B*AR,S,S)
    const int* __restrict__ qmask,      // (B,S)
    float* __restrict__ out_permu,      // (B*AR,S,S)
    float* __restrict__ out_macti,      // (B*AR,S,S)
    float* __restrict__ out_lognom)     // (B*AR)
{
  extern __shared__ float sm[];
  float* L     = sm;                         // 256*257
  float* cval  = sm + SS * PITCH;            // 256
  int*   crow  = (int*)(cval + SS);          // 256
  int*   ccol  = crow + SS;                  // 256
  int*   rdone = ccol + SS;                  // 256
  int*   cdone = rdone + SS;                 // 256
  int*   assig = cdone + SS;                 // 256

  const int r = blockIdx.x, t = threadIdx.x;
  const int b = r >> 2;
  const float* A = act    + (size_t)b * SS * SS;
  const float* G = gumbel + (size_t)r * SS * SS;
  const int mcol = qmask[b * SS + t] != 0;

#if HAVE_TDM
  // Stage activations tile via TDM (pitch-257 pad in hardware), then add the
  // masked Gumbel noise in place and emit masked_refinement / zeroed permu.
  if (threadIdx.x < 32) {
    tdm_load_256x256_padded(A, __builtin_amdgcn_groupstaticsize());
    __builtin_amdgcn_s_wait_tensorcnt(0);
  }
  rdone[t] = 0; cdone[t] = 0;
  __syncthreads();
  for (int i = 0; i < SS; ++i) {
    const int mi = qmask[b * SS + i] != 0;
    const float a = L[i * PITCH + t];
    const int vm = mi && mcol;
    if (vm) L[i * PITCH + t] = a + G[i * SS + t];
    out_macti[(size_t)r * SS * SS + i * SS + t] = vm ? a : 0.0f;
    out_permu[(size_t)r * SS * SS + i * SS + t] = 0.0f;
  }
  __syncthreads();
#else
  for (int i = 0; i < SS; ++i) {
    const int mi = qmask[b * SS + i] != 0;
    const float a = A[i * SS + t];
    const int vm = mi && mcol;
    L[i * PITCH + t] = a + (vm ? G[i * SS + t] : 0.0f);
    out_macti[(size_t)r * SS * SS + i * SS + t] = vm ? a : 0.0f;
    out_permu[(size_t)r * SS * SS + i * SS + t] = 0.0f;
  }
  rdone[t] = 0; cdone[t] = 0;
  __syncthreads();
#endif

  // 200 Sinkhorn iterations, entirely LDS-resident (zero HBM traffic)
  for (int it = 0; it < SINK_ITERS; ++it) sinkhorn_norm(L, t);

  // ds = exp(L)
  for (int i = 0; i < SS; ++i) L[i * PITCH + t] = __expf(L[i * PITCH + t]);
  __syncthreads();

  // Greedy assignment: 256 steps of global argmax (first-flat-index ties)
  for (int step = 0; step < SS; ++step) {
    float bv = -1.0f; int bj = -1;
    if (!rdone[t]) {
      for (int j = 0; j < SS; ++j) {
        if (cdone[j]) continue;
        const float v = L[t * PITCH + j];
        if (v > bv) { bv = v; bj = j; }   // strict > keeps lowest j on ties
      }
    }
    cval[t] = (bj >= 0) ? bv : -1e30f;
    crow[t] = t; ccol[t] = bj;
    __syncthreads();
    for (int off = 128; off > 0; off >>= 1) {
      if (t < off) {
        const bool take = (cval[t + off] > cval[t]) ||
                          (cval[t + off] == cval[t] && crow[t + off] < crow[t]);
        if (take) { cval[t] = cval[t+off]; crow[t] = crow[t+off]; ccol[t] = ccol[t+off]; }
      }
      __syncthreads();
    }
    if (t == 0) {
      const int rr = crow[0], cc = ccol[0];
      assig[rr] = cc; rdone[rr] = 1; cdone[cc] = 1;
    }
    __syncthreads();
  }

  // sample_permu one-hot + log_nominator
  const int aj = assig[t];
  out_permu[(size_t)r * SS * SS + t * SS + aj] = 1.0f;
  const int vm = (qmask[b * SS + t] != 0) && (qmask[b * SS + aj] != 0);
  const float contrib = vm ? A[t * SS + aj] : 0.0f;
  const float ln = block_sum(contrib, cval, t);
  if (t == 0) out_lognom[r] = ln;
}

// ---------------------------------------------------------------------------
extern "C" void kernel_launch(void* const* d_in, const int* in_sizes, int n_in,
                              void* d_out, int out_size, void* d_ws, size_t ws_size,
                              hipStream_t stream) {
  (void)in_sizes; (void)n_in; (void)out_size; (void)ws_size;
  const float* queries = (const float*)d_in[0];  // (32,256,512)
  const float* w1      = (const float*)d_in[1];  // (512,256)
  const float* b1      = (const float*)d_in[2];  // (256)
  const float* w2      = (const float*)d_in[3];  // (256,1024)
  const float* b2      = (const float*)d_in[4];  // (1024)
  const float* gumbel  = (const float*)d_in[5];  // (128,256,256)
  const int*   qmask   = (const int*)d_in[6];    // (32,256)
  const int*   pdone   = (const int*)d_in[7];    // scalar
  // d_in[8] = action_refinement (fixed 4)

  float* ws   = (float*)d_ws;
  float* h    = ws;                               // 8192*256
  float* acti = h    + (size_t)8192 * 256;        // 8192*1024
  float* act  = acti + (size_t)8192 * 1024;       // 32*256*256

  float* out        = (float*)d_out;
  float* out_permu  = out;                                    // 128*256*256
  float* out_macti  = out_permu + (size_t)128 * 256 * 256;    // 128*256*256
  float* out_kl     = out_macti + (size_t)128 * 256 * 256;    // 128
  float* out_lognom = out_kl + 128;                           // 128
  float* out_lnc    = out_lognom + 128;                       // 128

  // MLP: h = relu(queries@w1 + b1) ; acti = h@w2 + b2
  gemm_bias_act<<<1024, 256, 0, stream>>>(queries, w1, b1, h, 8192, 256, 512, 1);
  gemm_bias_act<<<4096, 256, 0, stream>>>(h, w2, b2, acti, 8192, 1024, 256, 0);

  // activations (masked multi-head scores)
  scores_kernel<<<1024, 256, 0, stream>>>(acti, qmask, pdone, act);

  // KL divergence per batch
  kl_kernel<<<32, 256, 0, stream>>>(act, qmask, out_kl);

  // Bethe free energy -> log_normalize_const
  const size_t bethe_smem = (size_t)(SS * PITCH + SS) * sizeof(float);   // ~264 KB
  bethe_kernel<<<32, 256, bethe_smem, stream>>>(act, qmask, out_lnc);

  // Sinkhorn(200) + greedy assignment + outputs (one WGP per matrix)
  const size_t sink_smem = (size_t)SS * PITCH * sizeof(float)
                         + (size_t)6 * SS * sizeof(int);                 // ~269 KB
  sinkhorn_kernel<<<128, 256, sink_smem, stream>>>(act, gumbel, qmask,
                                                   out_permu, out_macti, out_lognom);
}